// SEALGNN_76587856822555
// MI455X (gfx1250) — compile-verified
//
#include <hip/hip_runtime.h>

#define NNODES 100000
#define NEDGES 3200000
#define HID 64
#define NGRAPH 128
#define APITCH 68   // float pitch: 16B-aligned rows, conflict-free frag reads

typedef __attribute__((ext_vector_type(2))) float v2f;
typedef __attribute__((ext_vector_type(8))) float v8f;

// ---------------- utility kernels ----------------
__global__ void k_zero(float* p, int n) {
    int i = blockIdx.x * blockDim.x + threadIdx.x;
    if (i < n) p[i] = 0.0f;
}

__global__ void k_deg(const int* __restrict__ dst, const float* __restrict__ ew,
                      float* __restrict__ deg) {
    int e = blockIdx.x * blockDim.x + threadIdx.x;
    if (e < NEDGES) atomicAdd(&deg[dst[e]], ew[e]);
}

__global__ void k_cnt(const int* __restrict__ batch, float* __restrict__ gcnt) {
    int i = blockIdx.x * blockDim.x + threadIdx.x;
    if (i < NNODES) atomicAdd(&gcnt[batch[i]], 1.0f);
}

__global__ void k_rsqrt(float* __restrict__ d, int n) {
    int i = blockIdx.x * blockDim.x + threadIdx.x;
    if (i < n) d[i] = rsqrtf(d[i] + 1.0f);
}

// ---------------- WMMA GEMM + fused GCN epilogue ----------------
// Computes h = A @ W, and agg = h*dinv^2 + bias (self-loop + bias init).
// MODE 0: A row r = emb[labels[r]] (embedding gather).
// MODE 1: A row r = relu(Ain[r])   (layer-2 activation fused on load).
template <int MODE>
__global__ __launch_bounds__(256) void k_gemm64(const float* __restrict__ Ain,
                                                const int* __restrict__ labels,
                                                const float* __restrict__ emb,
                                                const float* __restrict__ W,
                                                float* __restrict__ hOut,
                                                float* __restrict__ aggOut,
                                                const float* __restrict__ dinv,
                                                const float* __restrict__ bias,
                                                int nrows) {
    __shared__ float sW[64 * APITCH];
    __shared__ float sA[128 * APITCH];
    const int tid = threadIdx.x;
    const int rbase = blockIdx.x * 128;

    // Stage W (64x64) into LDS, float4 loads
    for (int i = tid; i < 64 * 16; i += 256) {
        int r = i >> 4, c4 = i & 15;
        float4 w4 = ((const float4*)W)[i];
        ((float4*)&sW[r * APITCH])[c4] = w4;
    }
    // Stage A (128 rows x 64) into LDS
    for (int i = tid; i < 128 * 16; i += 256) {
        int r = i >> 4, c4 = i & 15;
        int grow = rbase + r;
        float4 a4 = make_float4(0.f, 0.f, 0.f, 0.f);
        if (grow < nrows) {
            if (MODE == 0) {
                int lb = labels[grow];
                a4 = ((const float4*)(emb + (size_t)lb * 64))[c4];
            } else {
                a4 = ((const float4*)(Ain + (size_t)grow * 64))[c4];
                a4.x = fmaxf(a4.x, 0.f); a4.y = fmaxf(a4.y, 0.f);
                a4.z = fmaxf(a4.z, 0.f); a4.w = fmaxf(a4.w, 0.f);
            }
        }
        ((float4*)&sA[r * APITCH])[c4] = a4;
    }
    __syncthreads();

    const int lane = tid & 31;
    const int wv = tid >> 5;            // 8 waves, 16 rows each
    const int m = lane & 15;
    const int ksel = (lane >> 4) << 1;  // 0 for lanes 0-15, 2 for lanes 16-31

    v8f acc[4] = {};                    // 4 N-tiles of 16 cols
    const float* aRow = &sA[(wv * 16 + m) * APITCH];

    for (int k0 = 0; k0 < 64; k0 += 4) {
        v2f a;
        a.x = aRow[k0 + ksel];
        a.y = aRow[k0 + ksel + 1];
#pragma unroll
        for (int nt = 0; nt < 4; nt++) {
            v2f b;
            b.x = sW[(k0 + ksel) * APITCH + nt * 16 + m];
            b.y = sW[(k0 + ksel + 1) * APITCH + nt * 16 + m];
            acc[nt] = __builtin_amdgcn_wmma_f32_16x16x4_f32(
                false, a, false, b, (short)0, acc[nt], false, false);
        }
    }

    // Stage C back to LDS (conflict-free with pitch 68), then coalesced stores.
    __syncthreads();  // all waves done reading sA
    const int rloc = wv * 16 + ((lane >> 4) << 3);
#pragma unroll
    for (int nt = 0; nt < 4; nt++) {
#pragma unroll
        for (int j = 0; j < 8; j++) {
            sA[(rloc + j) * APITCH + nt * 16 + m] = acc[nt][j];
        }
    }
    __syncthreads();

    // Fused epilogue: h store + agg = h*dinv^2 + bias, coalesced float4.
    for (int i = tid; i < 128 * 16; i += 256) {
        int r = i >> 4, c4 = i & 15;
        int grow = rbase + r;
        if (grow < nrows) {
            float4 hv = ((const float4*)&sA[r * APITCH])[c4];
            ((float4*)hOut)[(size_t)grow * 16 + c4] = hv;
            float d = dinv[grow];
            float d2 = d * d;
            float4 bv = ((const float4*)bias)[c4];
            float4 o;
            o.x = hv.x * d2 + bv.x; o.y = hv.y * d2 + bv.y;
            o.z = hv.z * d2 + bv.z; o.w = hv.w * d2 + bv.w;
            ((float4*)aggOut)[(size_t)grow * 16 + c4] = o;
        }
    }
}

// ---------------- wave-cooperative edge scatter ----------------
__global__ __launch_bounds__(256) void k_scatter(const int* __restrict__ src,
                                                 const int* __restrict__ dst,
                                                 const float* __restrict__ ew,
                                                 const float* __restrict__ dinv,
                                                 const float* __restrict__ h,
                                                 float* __restrict__ agg) {
    const int lane = threadIdx.x & 31;
    const int wave = (blockIdx.x * blockDim.x + threadIdx.x) >> 5;
    const int nwaves = (gridDim.x * blockDim.x) >> 5;
    for (int e = wave; e < NEDGES; e += nwaves) {
        int s = src[e], d = dst[e];
        float norm = dinv[s] * ew[e] * dinv[d];
        float v0 = h[(size_t)s * 64 + lane];
        float v1 = h[(size_t)s * 64 + 32 + lane];
        atomicAdd(&agg[(size_t)d * 64 + lane], v0 * norm);
        atomicAdd(&agg[(size_t)d * 64 + 32 + lane], v1 * norm);
    }
}

// ---------------- relu + mean-pool accumulate ----------------
__global__ void k_pool(const float* __restrict__ agg, const int* __restrict__ batch,
                       float* __restrict__ gsum) {
    int i = blockIdx.x * blockDim.x + threadIdx.x;
    if (i >= NNODES * 64) return;
    int node = i >> 6, c = i & 63;
    float v = fmaxf(agg[i], 0.f);
    atomicAdd(&gsum[batch[node] * 64 + c], v);
}

// ---------------- head: g = gsum/cnt; relu(g@W3+b3) @ W4 + b4 ----------------
__global__ void k_head(const float* __restrict__ gsum, const float* __restrict__ gcnt,
                       const float* __restrict__ W3, const float* __restrict__ b3,
                       const float* __restrict__ W4, const float* __restrict__ b4,
                       float* __restrict__ out) {
    int g = blockIdx.x;
    int c = threadIdx.x;  // 64 threads
    __shared__ float gv[64];
    __shared__ float red[64];
    gv[c] = gsum[g * 64 + c] / fmaxf(gcnt[g], 1.0f);
    __syncthreads();
    float acc = b3[c];
    for (int k = 0; k < 64; k++) acc += gv[k] * W3[k * 64 + c];
    red[c] = fmaxf(acc, 0.f) * W4[c];
    __syncthreads();
    for (int s = 32; s > 0; s >>= 1) {
        if (c < s) red[c] += red[c + s];
        __syncthreads();
    }
    if (c == 0) out[g] = red[0] + b4[0];
}

extern "C" void kernel_launch(void* const* d_in, const int* in_sizes, int n_in,
                              void* d_out, int out_size, void* d_ws, size_t ws_size,
                              hipStream_t stream) {
    const int*   labels = (const int*)d_in[0];
    const int*   eidx   = (const int*)d_in[1];
    const float* ew     = (const float*)d_in[2];
    const int*   batch  = (const int*)d_in[3];
    const float* emb    = (const float*)d_in[4];
    const float* W1 = (const float*)d_in[5];  const float* b1 = (const float*)d_in[6];
    const float* W2 = (const float*)d_in[7];  const float* b2 = (const float*)d_in[8];
    const float* W3 = (const float*)d_in[9];  const float* b3 = (const float*)d_in[10];
    const float* W4 = (const float*)d_in[11]; const float* b4 = (const float*)d_in[12];
    const int* esrc = eidx;
    const int* edst = eidx + NEDGES;

    float* ws   = (float*)d_ws;
    float* h    = ws;                             // N*64
    float* agg  = h    + (size_t)NNODES * 64;     // N*64
    float* dinv = agg  + (size_t)NNODES * 64;     // N   (holds deg, then dinv)
    float* gsum = dinv + NNODES;                  // G*64
    float* gcnt = gsum + NGRAPH * 64;             // G
    float* out  = (float*)d_out;

    k_zero<<<(NNODES + 255) / 256, 256, 0, stream>>>(dinv, NNODES);
    k_zero<<<(NGRAPH * 65 + 255) / 256, 256, 0, stream>>>(gsum, NGRAPH * 65);
    k_deg<<<(NEDGES + 255) / 256, 256, 0, stream>>>(edst, ew, dinv);
    k_cnt<<<(NNODES + 255) / 256, 256, 0, stream>>>(batch, gcnt);
    k_rsqrt<<<(NNODES + 255) / 256, 256, 0, stream>>>(dinv, NNODES);

    const int gblocks = (NNODES + 127) / 128;
    // Layer 1: h = emb[labels] @ W1 ; agg = h*dinv^2 + b1 ; scatter
    k_gemm64<0><<<gblocks, 256, 0, stream>>>(nullptr, labels, emb, W1,
                                             h, agg, dinv, b1, NNODES);
    k_scatter<<<4096, 256, 0, stream>>>(esrc, edst, ew, dinv, h, agg);
    // Layer 2: h = relu(agg) @ W2 ; agg = h*dinv^2 + b2 ; scatter
    k_gemm64<1><<<gblocks, 256, 0, stream>>>(agg, nullptr, nullptr, W2,
                                             h, agg, dinv, b2, NNODES);
    k_scatter<<<4096, 256, 0, stream>>>(esrc, edst, ew, dinv, h, agg);
    // Pool + head
    k_pool<<<(NNODES * 64 + 255) / 256, 256, 0, stream>>>(agg, batch, gsum);
    k_head<<<NGRAPH, 64, 0, stream>>>(gsum, gcnt, W3, b3, W4, b4, out);
}